// Propagation_78383153152422
// MI455X (gfx1250) — compile-verified
//
#include <hip/hip_runtime.h>
#include <stdint.h>

#define NEIGH 16
#define Bb 4
#define Dd 32
#define Hh 512
#define Ww 640

#define TX 32
#define TY 8
#define HALO_LO 7
#define HALO_HI 5
#define ROWS (TY + HALO_LO + HALO_HI)   // 20
#define COLS (TX + HALO_LO + HALO_HI)   // 44
#define COLSP 45                        // LDS pitch (odd -> spread banks)

typedef float v2f __attribute__((ext_vector_type(2)));
typedef float v8f __attribute__((ext_vector_type(8)));

__device__ __forceinline__ int refl(int i, int n) {
    i = (i < 0) ? -i : i;
    i = (i >= n) ? (2 * n - 2 - i) : i;
    return i;
}

// ---------------------------------------------------------------------------
// Kernel 1: stream depth_sample -> out channels [0,32) with (x<=0 ? 1 : x).
// Pure bandwidth, float4-vectorized.
// ---------------------------------------------------------------------------
__global__ __launch_bounds__(256) void clamp_copy_kernel(
    const float4* __restrict__ in4, float* __restrict__ out)
{
    const size_t HW   = (size_t)Hh * Ww;
    const size_t CIN  = (size_t)Dd * HW;             // elems per batch (input)
    const size_t tot4 = (size_t)Bb * CIN / 4;
    size_t i4 = (size_t)blockIdx.x * blockDim.x + threadIdx.x;
    if (i4 >= tot4) return;
    size_t i   = i4 * 4;
    size_t b   = i / CIN;
    size_t rem = i - b * CIN;
    float4 v = in4[i4];
    float4 w;
    w.x = (v.x <= 0.f) ? 1.f : v.x;
    w.y = (v.y <= 0.f) ? 1.f : v.y;
    w.z = (v.z <= 0.f) ? 1.f : v.z;
    w.w = (v.w <= 0.f) ? 1.f : v.w;
    float4* o4 = (float4*)(out + b * 48 * HW + rem);
    *o4 = w;
}

// ---------------------------------------------------------------------------
// Kernel 2: stencil + WMMA affine -> out channels [32,48) with clamp.
//   - center plane tile (reflect-padded) staged via global_load_async_to_lds
//   - per-pixel ref = 0.5 * sum_s(off_hi[s]*g[s] + off_lo[s]*g1[s])
//   - prop[c,p] = alpha[c]*ref[p] + beta_eff[c] via V_WMMA_F32_16X16X4_F32
// ---------------------------------------------------------------------------
__global__ __launch_bounds__(256) void prop_kernel(
    const float* __restrict__ depth, const float* __restrict__ off,
    const float* __restrict__ conv_w, const float* __restrict__ bn_gamma,
    const float* __restrict__ bn_beta, const float* __restrict__ bn_mean,
    const float* __restrict__ bn_var, float* __restrict__ out)
{
    __shared__ float tile[ROWS * COLSP];

    const int b    = blockIdx.z;
    const int x0   = blockIdx.x * TX;
    const int y0   = blockIdx.y * TY;
    const int tid  = threadIdx.x;
    const int lane = tid & 31;
    const int wrow = tid >> 5;        // 0..7 : row within the tile for this wave

    const size_t HW = (size_t)Hh * Ww;

    // ---- async stage: reflect-padded center-plane tile -> LDS --------------
    const float* center = depth + ((size_t)b * Dd + Dd / 2) * HW;
    const uint32_t lds0 = (uint32_t)(uintptr_t)&tile[0];
    for (int e = tid; e < ROWS * COLS; e += 256) {
        int r  = e / COLS;
        int c  = e - r * COLS;
        int gy = refl(y0 - HALO_LO + r, Hh);
        int gx = refl(x0 - HALO_LO + c, Ww);
        const float* gp = center + (size_t)gy * Ww + gx;
        uint32_t laddr  = lds0 + (uint32_t)(r * COLSP + c) * 4u;
        asm volatile("global_load_async_to_lds_b32 %0, %1, off"
                     :: "v"(laddr), "v"(gp) : "memory");
    }
    asm volatile("s_wait_asynccnt 0" ::: "memory");
    __syncthreads();

    // ---- per-pixel 32-tap dot product -------------------------------------
    const int y  = y0 + wrow;
    const int x  = x0 + lane;
    const int ty = wrow + HALO_LO;
    const int tx = lane + HALO_LO;
    const float* offp = off + ((size_t)b * 2 * NEIGH) * HW + (size_t)y * Ww + x;

    float acc = 0.f;
#pragma unroll
    for (int s = 0; s < NEIGH; ++s) {
        const int dy  = (s >> 2) * 2 - 3;   // {-3,-1,1,3}
        const int dx  = (s & 3) * 2 - 3;
        const int dy1 = (s >> 2) * 4 - 7;   // {-7,-3,1,5}
        const int dx1 = (s & 3) * 4 - 7;
        float g   = tile[(ty + dy)  * COLSP + (tx + dx)];
        float g1  = tile[(ty + dy1) * COLSP + (tx + dx1)];
        float olo = offp[(size_t)s * HW];            // offset[:, s]
        float ohi = offp[(size_t)(NEIGH + s) * HW];  // offset[:, 16+s]
        acc = fmaf(ohi, g, acc);
        acc = fmaf(olo, g1, acc);
    }
    const float ref = 0.5f * acc;

    // ---- channel affine via WMMA: D[c,p] = alpha[c]*ref[p] + beta_eff[c] ---
    // A (16x4 f32): lanes 0-15 hold M=lane, VGPR0=K0, VGPR1=K1; lanes 16-31
    // hold K2,K3 (zeroed). B (4x16): symmetric K-split; unused rows zeroed.
    const int  cidx = lane & 15;
    const bool lo   = lane < 16;
    float scale = bn_gamma[cidx] * rsqrtf(bn_var[cidx] + 1e-5f);
    v2f A;
    A.x = lo ? fmaf(conv_w[cidx], scale, 1.0f) : 0.f;         // alpha
    A.y = lo ? (bn_beta[cidx] - bn_mean[cidx] * scale) : 0.f; // beta_eff

    // second WMMA needs refs of pixels 16..31 in lanes 0..15 (SWAPX16)
    float ref_sw = __int_as_float(
        __builtin_amdgcn_ds_swizzle(__float_as_int(ref), 0x401F));

    v2f B0; B0.x = lo ? ref    : 0.f; B0.y = lo ? 1.0f : 0.f;
    v2f B1; B1.x = lo ? ref_sw : 0.f; B1.y = B0.y;
    v8f c0 = {};
    v8f d0 = __builtin_amdgcn_wmma_f32_16x16x4_f32(
        false, A, false, B0, (short)0, c0, false, false);
    v8f d1 = __builtin_amdgcn_wmma_f32_16x16x4_f32(
        false, A, false, B1, (short)0, c0, false, false);

    // ---- clamp + store channels 32..47 ------------------------------------
    // D layout: VGPR r, lane L -> channel (r + 8*(L>=16)), pixel (L&15)
    float* outp = out + ((size_t)b * 48 + 32) * HW + (size_t)y * Ww + x0;
    const int n     = lane & 15;
    const int mbase = (lane >> 4) << 3;
#pragma unroll
    for (int r = 0; r < 8; ++r) {
        float v0 = d0[r]; v0 = (v0 <= 0.f) ? 1.f : v0;
        float v1 = d1[r]; v1 = (v1 <= 0.f) ? 1.f : v1;
        outp[(size_t)(mbase + r) * HW + n]      = v0;
        outp[(size_t)(mbase + r) * HW + 16 + n] = v1;
    }
}

extern "C" void kernel_launch(void* const* d_in, const int* in_sizes, int n_in,
                              void* d_out, int out_size, void* d_ws, size_t ws_size,
                              hipStream_t stream) {
    (void)in_sizes; (void)n_in; (void)out_size; (void)d_ws; (void)ws_size;
    const float* depth  = (const float*)d_in[0];
    const float* off    = (const float*)d_in[1];
    const float* conv_w = (const float*)d_in[2];
    const float* gamma  = (const float*)d_in[3];
    const float* beta   = (const float*)d_in[4];
    const float* mean   = (const float*)d_in[5];
    const float* var    = (const float*)d_in[6];
    float* out = (float*)d_out;

    const size_t HW   = (size_t)Hh * Ww;
    const size_t tot4 = (size_t)Bb * Dd * HW / 4;
    dim3 gridC((unsigned)((tot4 + 255) / 256));
    clamp_copy_kernel<<<gridC, 256, 0, stream>>>((const float4*)depth, out);

    dim3 gridP(Ww / TX, Hh / TY, Bb);   // 20 x 64 x 4
    prop_kernel<<<gridP, 256, 0, stream>>>(depth, off, conv_w, gamma, beta,
                                           mean, var, out);
}